// ARMAConv_58145267253837
// MI455X (gfx1250) — compile-verified
//
#include <hip/hip_runtime.h>

typedef float v2f __attribute__((ext_vector_type(2)));
typedef float v8f __attribute__((ext_vector_type(8)));

#define D 128
#define KSTACK 3
#define LDS_STRIDE 132   // 16B-aligned rows for ds_store_b128; conflict-free frag reads

// ---------------------------------------------------------------- utilities
__global__ void zero_f4_kernel(float4* __restrict__ p, int n4) {
  int i = blockIdx.x * blockDim.x + threadIdx.x;
  if (i < n4) p[i] = make_float4(0.f, 0.f, 0.f, 0.f);
}

__global__ void deg_kernel(const int* __restrict__ dst, float* __restrict__ deg, int nE) {
  int i = blockIdx.x * blockDim.x + threadIdx.x;
  if (i < nE) unsafeAtomicAdd(&deg[dst[i]], 1.0f);
}

__global__ void norm_kernel(float* __restrict__ d, int n) {
  int i = blockIdx.x * blockDim.x + threadIdx.x;
  if (i < n) d[i] = rsqrtf(fmaxf(d[i], 1.0f));
}

// ------------------------------------------------------------------- SpMM
// one wave32 per edge: agg[dst] += h[src] * norm[src]   (norm[dst] folded
// into the consumer GEMM's A load). Each lane owns a float4 (32*4 = 128).
__global__ void spmm_scatter_kernel(const float* __restrict__ h,
                                    const int* __restrict__ src,
                                    const int* __restrict__ dst,
                                    const float* __restrict__ nrm,
                                    float* __restrict__ agg, int nE) {
  const int wid  = threadIdx.x >> 5;
  const int lane = threadIdx.x & 31;
  const int e = blockIdx.x * (blockDim.x >> 5) + wid;
  if (e >= nE) return;
  const int s = src[e];
  const int d = dst[e];
  const float ns = nrm[s];
  const float4 v4 = *reinterpret_cast<const float4*>(h + (size_t)s * D + lane * 4);
  float* base = agg + (size_t)d * D + lane * 4;
  unsafeAtomicAdd(base + 0, v4.x * ns);
  unsafeAtomicAdd(base + 1, v4.y * ns);
  unsafeAtomicAdd(base + 2, v4.z * ns);
  unsafeAtomicAdd(base + 3, v4.w * ns);
}

// ------------------------------------------------------------------- GEMM
// out[row,col] = epilogue( (diag(nrm)*A) @ W + bias + addm )
// Block = 256 threads = 8 waves; wave w owns output columns [16w, 16w+16).
// Each wave preloads its full 128x16 B slice into 64 VGPRs once, then
// grid-strides over 16-row A tiles. A tiles are staged in registers as
// float4 (global_load_b128), pipelined so the next tile's HBM loads are in
// flight during the current tile's 32 v_wmma_f32_16x16x4_f32 ops.
//
// WMMA f32 16x16x4 fragment layout (wave32):
//   A: lane L (m = L&15, half = L>>4): vgpr0 = A[m][k+2*half], vgpr1 = A[m][k+2*half+1]
//   B: lane L (n = L&15):              vgpr0 = W[k+2*half][n], vgpr1 = W[k+2*half+1][n]
//   C: vgpr r: lanes 0-15 -> M=r, lanes 16-31 -> M=r+8; N = L&15
template <bool HAS_NRM, bool HAS_ADD, bool RELU, bool ACCUM>
__global__ __launch_bounds__(256) void gemm_kernel(
    const float* __restrict__ A, const float* __restrict__ nrm,
    const float* __restrict__ W, const float* __restrict__ bias,
    const float* __restrict__ addm, float* __restrict__ out,
    int nRowTiles, float outScale) {
  __shared__ float As[16 * LDS_STRIDE];
  const int tid  = threadIdx.x;
  const int wid  = tid >> 5;
  const int lane = tid & 31;
  const int half = lane >> 4;
  const int lm   = lane & 15;
  const int colBase = wid * 16;

  // preload this wave's 128x16 slice of W into registers (64 VGPRs)
  v2f breg[32];
#pragma unroll
  for (int ks = 0; ks < 32; ++ks) {
    const int kk = ks * 4 + half * 2;
    breg[ks].x = W[(size_t)kk * D + colBase + lm];
    breg[ks].y = W[(size_t)(kk + 1) * D + colBase + lm];
  }
  const float bcol = bias[colBase + lm];

  // this thread's two float4 staging slots within a 16x128 tile
  const int idx0 = tid;              // slot 0..255
  const int idx1 = tid + 256;        // slot 256..511
  const int r0 = idx0 >> 5, c0 = (idx0 & 31) * 4;
  const int r1 = idx1 >> 5, c1 = (idx1 & 31) * 4;
  const float4* __restrict__ A4 = reinterpret_cast<const float4*>(A);

  int rt = blockIdx.x;
  if (rt >= nRowTiles) return;

  float4 s0, s1;
  {
    const int rowBase = rt * 16;
    s0 = A4[(size_t)(rowBase + r0) * 32 + (idx0 & 31)];
    s1 = A4[(size_t)(rowBase + r1) * 32 + (idx1 & 31)];
    if (HAS_NRM) {
      const float n0 = nrm[rowBase + r0], n1 = nrm[rowBase + r1];
      s0.x *= n0; s0.y *= n0; s0.z *= n0; s0.w *= n0;
      s1.x *= n1; s1.y *= n1; s1.z *= n1; s1.w *= n1;
    }
  }

  for (;;) {
    __syncthreads();   // previous tile's LDS reads are done
    *reinterpret_cast<float4*>(&As[r0 * LDS_STRIDE + c0]) = s0;
    *reinterpret_cast<float4*>(&As[r1 * LDS_STRIDE + c1]) = s1;

    const int rnext = rt + gridDim.x;          // block-uniform
    const bool more = rnext < nRowTiles;
    if (more) {                                // prefetch next tile into regs
      const int rowBase = rnext * 16;
      s0 = A4[(size_t)(rowBase + r0) * 32 + (idx0 & 31)];
      s1 = A4[(size_t)(rowBase + r1) * 32 + (idx1 & 31)];
      if (HAS_NRM) {
        const float n0 = nrm[rowBase + r0], n1 = nrm[rowBase + r1];
        s0.x *= n0; s0.y *= n0; s0.z *= n0; s0.w *= n0;
        s1.x *= n1; s1.y *= n1; s1.z *= n1; s1.w *= n1;
      }
    }
    __syncthreads();   // LDS tile visible

    v8f acc = {0.f, 0.f, 0.f, 0.f, 0.f, 0.f, 0.f, 0.f};
#pragma unroll
    for (int ks = 0; ks < 32; ++ks) {
      const int kk = ks * 4 + half * 2;
      v2f a;
      a.x = As[lm * LDS_STRIDE + kk];
      a.y = As[lm * LDS_STRIDE + kk + 1];
      acc = __builtin_amdgcn_wmma_f32_16x16x4_f32(
          /*neg_a=*/false, a, /*neg_b=*/false, breg[ks],
          /*c_mod=*/(short)0, acc, /*reuse_a=*/false, /*reuse_b=*/false);
    }

    // fused epilogue: + bias + addm, ReLU, scale, (accumulate|write)
    const int rowBase = rt * 16;
#pragma unroll
    for (int r = 0; r < 8; ++r) {
      const int row = rowBase + r + half * 8;
      const int col = colBase + lm;
      const size_t oi = (size_t)row * D + col;
      float vv = acc[r] + bcol;
      if (HAS_ADD) vv += addm[oi];
      if (RELU) vv = fmaxf(vv, 0.f);
      vv *= outScale;
      if (ACCUM) out[oi] += vv;
      else       out[oi]  = vv;
    }

    if (!more) break;
    rt = rnext;
  }
}

// ------------------------------------------------------------------- launch
extern "C" void kernel_launch(void* const* d_in, const int* in_sizes, int n_in,
                              void* d_out, int out_size, void* d_ws, size_t ws_size,
                              hipStream_t stream) {
  const float* feats = (const float*)d_in[0];
  const int*   src   = (const int*)d_in[1];
  const int*   dst   = (const int*)d_in[2];
  const float* w0    = (const float*)d_in[3];
  const float* b0    = (const float*)d_in[4];
  const float* w     = (const float*)d_in[5];
  const float* bw    = (const float*)d_in[6];
  const float* v     = (const float*)d_in[7];
  const float* bv    = (const float*)d_in[8];
  float* out = (float*)d_out;

  const int N = in_sizes[0] / D;   // 50000
  const int E = in_sizes[1];       // 800000
  const size_t M = (size_t)N * D;

  float* ws = (float*)d_ws;
  size_t off = 0;
  float* nrm  = ws + off; off += ((size_t)N + 255) & ~(size_t)255;
  float* agg0 = ws + off; off += M;
  float* fv   = ws + off; off += M;
  float* h1   = ws + off; off += M;
  float* agg1 = ws + off; off += M;

  const int nRowTiles  = N / 16;   // 3125
  const int gemmBlocks = 625;      // 5 tiles per block, uniform

  // degree -> norm = rsqrt(max(deg,1))
  zero_f4_kernel<<<(N / 4 + 255) / 256, 256, 0, stream>>>((float4*)nrm, N / 4);
  deg_kernel<<<(E + 255) / 256, 256, 0, stream>>>(dst, nrm, E);
  norm_kernel<<<(N + 255) / 256, 256, 0, stream>>>(nrm, N);

  // agg0 = scatter_add(feats[src] * norm[src])   (shared across all k)
  zero_f4_kernel<<<((int)(M / 4) + 255) / 256, 256, 0, stream>>>((float4*)agg0, (int)(M / 4));
  spmm_scatter_kernel<<<(E + 7) / 8, 256, 0, stream>>>(feats, src, dst, nrm, agg0, E);

  for (int k = 0; k < KSTACK; ++k) {
    // fv = feats @ v[k] + bv[k]          (reused at t=0 and t=1)
    gemm_kernel<false, false, false, false><<<gemmBlocks, 256, 0, stream>>>(
        feats, nullptr, v + (size_t)k * D * D, bv + (size_t)k * D,
        nullptr, fv, nRowTiles, 1.0f);
    // h1 = relu( (agg0*norm) @ w0[k] + b0[k] + fv )
    gemm_kernel<true, true, true, false><<<gemmBlocks, 256, 0, stream>>>(
        agg0, nrm, w0 + (size_t)k * D * D, b0 + (size_t)k * D,
        fv, h1, nRowTiles, 1.0f);
    // agg1 = scatter_add(h1[src] * norm[src])
    zero_f4_kernel<<<((int)(M / 4) + 255) / 256, 256, 0, stream>>>((float4*)agg1, (int)(M / 4));
    spmm_scatter_kernel<<<(E + 7) / 8, 256, 0, stream>>>(h1, src, dst, nrm, agg1, E);
    // out (+)= relu( (agg1*norm) @ w[k] + bw[k] + fv ) / K
    if (k == 0) {
      gemm_kernel<true, true, true, false><<<gemmBlocks, 256, 0, stream>>>(
          agg1, nrm, w + (size_t)k * D * D, bw + (size_t)k * D,
          fv, out, nRowTiles, 1.0f / (float)KSTACK);
    } else {
      gemm_kernel<true, true, true, true><<<gemmBlocks, 256, 0, stream>>>(
          agg1, nrm, w + (size_t)k * D * D, bw + (size_t)k * D,
          fv, out, nRowTiles, 1.0f / (float)KSTACK);
    }
  }
}